// Att_mlp_sigmod_88545045774670
// MI455X (gfx1250) — compile-verified
//
#include <hip/hip_runtime.h>
#include <hip/hip_bf16.h>
#include <math.h>

// Problem constants (from the reference):
#define GN 8192   // nodes
#define GD 256    // feature dim
#define GH 64     // MLP hidden
#define TPB 256
#define CHUNK 2048

typedef float v2f __attribute__((ext_vector_type(2)));
typedef float v8f __attribute__((ext_vector_type(8)));

// ---------------------------------------------------------------------------
// Kernel 1: e = sigmoid(relu(h @ W1 + b1) @ W2 + b2), one wave per 16 rows.
// Uses V_WMMA_F32_16X16X4_F32 (full fp32 precision) per CDNA5 ISA layouts:
//   A (16x4 f32):  lanes 0-15 -> M=lane, VGPR0/1 = K=k,k+1 ; lanes 16-31 -> K=k+2,k+3
//   B (4x16 f32):  lanes 0-15 -> N=lane, VGPR0/1 = K=k,k+1 ; lanes 16-31 -> K=k+2,k+3
//   C/D (16x16 f32, 8 VGPRs): element r = C[M = r + 8*(lane>>4), N = lane&15]
// ---------------------------------------------------------------------------
__global__ __launch_bounds__(32) void mlp_gate_wmma(
    const float* __restrict__ h, const float* __restrict__ W1,
    const float* __restrict__ b1, const float* __restrict__ W2,
    const float* __restrict__ b2, float* __restrict__ e_out)
{
  const int row0 = blockIdx.x << 4;          // 16 rows per wave
  const int lane = threadIdx.x & 31;
  const int half = lane >> 4;                // 0 | 1
  const int m    = lane & 15;

  v8f acc[4] = {};                           // 4 N-tiles (H=64 = 4*16), each 16x16 fp32

#if defined(__gfx1250__) && __has_builtin(__builtin_amdgcn_wmma_f32_16x16x4_f32)
  const float* __restrict__ hrow = h + (size_t)(row0 + m) * GD;
  for (int k = 0; k < GD; k += 4) {
    const int ka = k + (half << 1);
    v2f a;
    a.x = hrow[ka];
    a.y = hrow[ka + 1];
    const float* __restrict__ wr = W1 + (size_t)ka * GH;   // W1 row-major [256,64]
#pragma unroll
    for (int t = 0; t < 4; ++t) {
      v2f b;
      b.x = wr[t * 16 + m];                  // W1[ka  , t*16+m]
      b.y = wr[GH + t * 16 + m];             // W1[ka+1, t*16+m]
      acc[t] = __builtin_amdgcn_wmma_f32_16x16x4_f32(
          /*neg_a=*/false, a, /*neg_b=*/false, b,
          /*c_mod=*/(short)0, acc[t], /*reuse_a=*/false, /*reuse_b=*/false);
    }
  }
#else
  // Scalar fallback producing the exact same per-lane C-tile layout.
  for (int k = 0; k < GD; ++k) {
    float wv[4];
#pragma unroll
    for (int t = 0; t < 4; ++t) wv[t] = W1[(size_t)k * GH + t * 16 + m];
#pragma unroll
    for (int r = 0; r < 8; ++r) {
      float hv = h[(size_t)(row0 + r + (half << 3)) * GD + k];
#pragma unroll
      for (int t = 0; t < 4; ++t) acc[t][r] = fmaf(hv, wv[t], acc[t][r]);
    }
  }
#endif

  // Epilogue: relu(+b1), dot with W2 (reduce across the 64 hidden cols), +b2, sigmoid.
  const float vb2 = b2[0];
  float w2c[4], b1c[4];
#pragma unroll
  for (int t = 0; t < 4; ++t) {
    const int col = t * 16 + m;
    w2c[t] = W2[col];
    b1c[t] = b1[col];
  }

#pragma unroll
  for (int r = 0; r < 8; ++r) {
    float s = 0.f;
#pragma unroll
    for (int t = 0; t < 4; ++t) {
      float hid = acc[t][r] + b1c[t];
      hid = hid > 0.f ? hid : 0.f;           // relu
      s = fmaf(hid, w2c[t], s);
    }
    // Reduce over the 16 lanes of this half (xor masks 1/2/4/8 stay in-half on wave32).
    s += __shfl_xor(s, 1, 32);
    s += __shfl_xor(s, 2, 32);
    s += __shfl_xor(s, 4, 32);
    s += __shfl_xor(s, 8, 32);
    if (m == 0) {
      const float x = s + vb2;
      e_out[row0 + r + (half << 3)] = 1.0f / (1.0f + __expf(-x));
    }
  }
}

// ---------------------------------------------------------------------------
// Kernel 2: h_out[i,:] = sum_{j : A[i,j] > 0} e[j] * h[j,:]
// One block per row. Streams the 32KB adjacency row (the 256MB HBM-bound part),
// compacts nonzero column indices DETERMINISTICALLY (ascending j) into LDS via
// wave prefix-scan, then each thread owns one of the 256 output dims.
// ---------------------------------------------------------------------------
__global__ __launch_bounds__(TPB) void spmm_gate(
    const float* __restrict__ A, const float* __restrict__ h,
    const float* __restrict__ e, float* __restrict__ h_out)
{
  __shared__ int   s_idx[CHUNK];
  __shared__ float s_val[CHUNK];
  __shared__ int   s_wsum[TPB / 32];

  const int row  = blockIdx.x;
  const int tid  = threadIdx.x;
  const int lane = tid & 31;
  const int wid  = tid >> 5;

  const float* __restrict__ arow = A + (size_t)row * GN;
  float acc = 0.f;

  for (int c0 = 0; c0 < GN; c0 += CHUNK) {
    // Pull the next chunk of the adjacency row toward the caches while we scan.
    if (c0 + CHUNK < GN)
      __builtin_prefetch(arow + c0 + CHUNK + tid * 8, 0, 1);

    const int base = c0 + tid * 8;           // 8 contiguous cols per thread
    const float4* p = (const float4*)(arow + base);
    const float4 v0 = p[0];
    const float4 v1 = p[1];
    const float vals[8] = {v0.x, v0.y, v0.z, v0.w, v1.x, v1.y, v1.z, v1.w};

    unsigned mask = 0u;
#pragma unroll
    for (int q = 0; q < 8; ++q) mask |= (vals[q] > 0.f) ? (1u << q) : 0u;
    const int cnt = __popc(mask);

    // Wave32 inclusive prefix scan of nonzero counts (keeps column order).
    int incl = cnt;
#pragma unroll
    for (int off = 1; off < 32; off <<= 1) {
      int y = __shfl_up(incl, off, 32);
      if (lane >= off) incl += y;
    }
    if (lane == 31) s_wsum[wid] = incl;
    __syncthreads();

    int off0  = incl - cnt;                  // exclusive offset within wave
    int total = 0;
#pragma unroll
    for (int w = 0; w < TPB / 32; ++w) {
      const int t = s_wsum[w];
      if (w < wid) off0 += t;
      total += t;
    }

    int slot = off0;
#pragma unroll
    for (int q = 0; q < 8; ++q) {
      if (mask & (1u << q)) {
        const int j = base + q;
        s_idx[slot] = j;
        s_val[slot] = e[j];
        ++slot;
      }
    }
    __syncthreads();

    // Accumulate: thread `tid` owns output dim `tid`; h-row reads are fully
    // coalesced 1KB bursts and hit L2 (h is only 8MB vs 192MB L2).
    for (int t = 0; t < total; ++t) {
      const int j = s_idx[t];
      acc = fmaf(s_val[t], h[(size_t)j * GD + tid], acc);
    }
    __syncthreads();
  }

  h_out[(size_t)row * GD + tid] = acc;
}

// ---------------------------------------------------------------------------
extern "C" void kernel_launch(void* const* d_in, const int* in_sizes, int n_in,
                              void* d_out, int out_size, void* d_ws, size_t ws_size,
                              hipStream_t stream) {
  const float* A  = (const float*)d_in[0];   // graph_info [N,N]
  const float* h  = (const float*)d_in[1];   // [N,D]
  const float* W1 = (const float*)d_in[2];   // [D,H]
  const float* b1 = (const float*)d_in[3];   // [H]
  const float* W2 = (const float*)d_in[4];   // [H,1]
  const float* b2 = (const float*)d_in[5];   // [1]

  float* out   = (float*)d_out;
  float* h_out = out;                        // [N,D] flat, first output
  float* e_out = out + (size_t)GN * GD;      // [N,1] flat, second output

  mlp_gate_wmma<<<GN / 16, 32, 0, stream>>>(h, W1, b1, W2, b2, e_out);
  spmm_gate<<<GN, TPB, 0, stream>>>(A, h, e_out, h_out);
}